// SeqLinkAttention_53008486367489
// MI455X (gfx1250) — compile-verified
//
#include <hip/hip_runtime.h>
#include <math.h>

typedef __attribute__((ext_vector_type(2))) float v2f;
typedef __attribute__((ext_vector_type(8))) float v8f;

#define Bn 512
#define Tn 2048
#define Dn 64
#define Hn 32
#define APAD 68   // padded A-tile row stride (floats) -> conflict-free ds_load_b64
#define TPW 4     // tiles per wave in k_fused (double-buffered)

// --- CDNA5 async global->LDS path (guarded; fallback = register pipeline) ---
#if defined(__has_builtin)
#  if __has_builtin(__builtin_amdgcn_global_load_async_to_lds_b128)
#    define HAVE_ASYNC_LDS 1
#  endif
#endif
#ifndef HAVE_ASYNC_LDS
#  define HAVE_ASYNC_LDS 0
#endif

#if HAVE_ASYNC_LDS
// Builtin expects: (int4 addrspace(1)*, int4 addrspace(3)*, imm, imm)
typedef int v4i_vs __attribute__((vector_size(16)));
typedef __attribute__((address_space(1))) v4i_vs gv4i;  // global int4
typedef __attribute__((address_space(3))) v4i_vs lv4i;  // LDS int4
#endif

#if defined(__has_builtin)
#  if __has_builtin(__builtin_amdgcn_s_wait_asynccnt)
#    define WAIT_ASYNC(n) __builtin_amdgcn_s_wait_asynccnt(n)
#  endif
#endif
#ifndef WAIT_ASYNC
#  define WAIT_ASYNC(n) asm volatile("s_wait_asynccnt " #n ::: "memory")
#endif

// ---------------------------------------------------------------------------
// Kernel 1: u_repr[b,d] = mean_t latent_ys[b,t,d].  512 blocks x 256 threads.
// ---------------------------------------------------------------------------
__global__ __launch_bounds__(256) void k_mean(const float* __restrict__ X,
                                              float* __restrict__ u) {
  const int b   = blockIdx.x;
  const int tid = threadIdx.x;
  const int q   = tid & 15;   // float4 column 0..15
  const int g   = tid >> 4;   // t-group 0..15
  const float4* Xb = (const float4*)(X + (size_t)b * Tn * Dn);
  float4 acc = {0.f, 0.f, 0.f, 0.f};
  for (int t = g; t < Tn; t += 16) {
    float4 v = Xb[t * 16 + q];
    acc.x += v.x; acc.y += v.y; acc.z += v.z; acc.w += v.w;
  }
  __shared__ float4 red[256];
  red[tid] = acc;
  __syncthreads();
  if (tid < 16) {
    float4 s = red[tid];
    for (int gg = 1; gg < 16; ++gg) {
      float4 v = red[gg * 16 + tid];
      s.x += v.x; s.y += v.y; s.z += v.z; s.w += v.w;
    }
    const float inv = 1.f / (float)Tn;
    float4 o = {s.x * inv, s.y * inv, s.z * inv, s.w * inv};
    ((float4*)u)[b * 16 + tid] = o;
  }
}

// ---------------------------------------------------------------------------
// Kernel 2: per-row softmax factorization + level thresholds.
// weights[i,j] = e[j] / (Z - e[i]),  e = exp(su - max),  su[j] = u[j] . v,
// v[d] = sum_h Wu[d,h] * Ws[H+h].   One block, 512 threads (thread = row).
// ---------------------------------------------------------------------------
__global__ __launch_bounds__(512) void k_pairwise(
    const float* __restrict__ u, const float* __restrict__ Wu,
    const float* __restrict__ Ws, float* __restrict__ e_out,
    float* __restrict__ mean0, float* __restrict__ mean1,
    float* __restrict__ invD, float* __restrict__ lc0,
    float* __restrict__ lc1, float* __restrict__ ac, int* __restrict__ fb) {
  __shared__ float v[Dn];
  __shared__ float sh_e[Bn];
  __shared__ float red[Bn];
  __shared__ int s_i1, s_i2;
  const int tid = threadIdx.x;

  if (tid < Dn) {
    float s = 0.f;
    for (int h = 0; h < Hn; ++h) s += Wu[tid * Hn + h] * Ws[Hn + h];
    v[tid] = s;
  }
  __syncthreads();

  float su = 0.f;
  for (int d = 0; d < Dn; ++d) su += u[tid * Dn + d] * v[d];

  red[tid] = su;
  __syncthreads();
  for (int s = Bn / 2; s > 0; s >>= 1) {
    if (tid < s) red[tid] = fmaxf(red[tid], red[tid + s]);
    __syncthreads();
  }
  const float m = red[0];
  __syncthreads();

  const float ej = expf(su - m);
  sh_e[tid] = ej;
  e_out[tid] = ej;
  red[tid] = ej;
  __syncthreads();
  for (int s = Bn / 2; s > 0; s >>= 1) {
    if (tid < s) red[tid] += red[tid + s];
    __syncthreads();
  }
  const float Z = red[0];
  __syncthreads();

  if (tid == 0) {  // top-2 of e[] (fallback index = argmax over j != i)
    float b1 = -1e30f, b2 = -1e30f;
    int i1 = 0, i2 = 0;
    for (int j = 0; j < Bn; ++j) {
      float x = sh_e[j];
      if (x > b1) { b2 = b1; i2 = i1; b1 = x; i1 = j; }
      else if (x > b2) { b2 = x; i2 = j; }
    }
    s_i1 = i1; s_i2 = i2;
  }
  __syncthreads();

  const int i = tid;
  const float iDi = 1.f / (Z - sh_e[i]);

  float asum = 0.f;
  for (int j = 0; j < Bn; ++j) { if (j == i) continue; asum += sh_e[j] * iDi; }
  const float m0 = asum / (float)(Bn - 1);

  int l0 = 0, cnt1 = 0;
  float asum1 = 0.f;
  for (int j = 0; j < Bn; ++j) {
    if (j == i) continue;
    float w = sh_e[j] * iDi;
    if (w <= m0) l0++; else { asum1 += w; cnt1++; }
  }
  const float m1v = asum1 / fmaxf((float)cnt1, 1.f);

  int l1 = 0;
  for (int j = 0; j < Bn; ++j) {
    if (j == i) continue;
    float w = sh_e[j] * iDi;
    if (w > m0 && w <= m1v) l1++;
  }

  mean0[i] = m0;
  mean1[i] = m1v;
  invD[i]  = iDi;
  lc0[i] = (float)l0;
  lc1[i] = (float)l1;
  ac[i]  = (float)(cnt1 - l1);
  fb[i]  = (i == s_i1) ? s_i2 : s_i1;
}

// ---------------------------------------------------------------------------
// Kernel 3: level means -> cross -> cbias[b,:] = cross @ Wf[D:] + bf.
// 512 blocks (row i) x 64 threads (dim d).
// ---------------------------------------------------------------------------
__global__ __launch_bounds__(64) void k_levels(
    const float* __restrict__ u, const float* __restrict__ e,
    const float* __restrict__ mean0, const float* __restrict__ mean1,
    const float* __restrict__ invD, const float* __restrict__ lc0,
    const float* __restrict__ lc1, const float* __restrict__ ac,
    const int* __restrict__ fb, const float* __restrict__ lw,
    const float* __restrict__ Wf, const float* __restrict__ bf,
    float* __restrict__ cbias) {
  const int i = blockIdx.x;
  const int d = threadIdx.x;
  __shared__ float sh_e[Bn];
  __shared__ float cross[Dn];
  for (int j = d; j < Bn; j += Dn) sh_e[j] = e[j];
  __syncthreads();

  const float iDi = invD[i];
  const float m0 = mean0[i], m1 = mean1[i];
  float a0 = 0.f, a1 = 0.f, a2 = 0.f;
  for (int j = 0; j < Bn; ++j) {
    if (j == i) continue;
    float w  = sh_e[j] * iDi;
    float uj = u[j * Dn + d];
    if (w <= m0)       a0 += uj;
    else if (w <= m1)  a1 += uj;
    else               a2 += uj;
  }
  const float c0 = (lc0[i] > 0.f) ? a0 / fmaxf(lc0[i], 1.f) : 0.f;
  const float c1 = (lc1[i] > 0.f) ? a1 / fmaxf(lc1[i], 1.f) : 0.f;
  const float c2 = (ac[i]  > 0.f) ? a2 / fmaxf(ac[i], 1.f) : u[fb[i] * Dn + d];
  cross[d] = lw[0] * c0 + lw[1] * c1 + lw[2] * c2;
  __syncthreads();

  float s = bf[d];
  for (int k = 0; k < Dn; ++k) s += cross[k] * Wf[(Dn + k) * Dn + d];
  cbias[i * Dn + d] = s;
}

// ---------------------------------------------------------------------------
// Kernel 4: out = LN( X @ (Wf1 + I) + cbias[b] ) via WMMA f32 16x16x4.
// One wave processes TPW consecutive 16x64 tiles, double-buffered A staging
// (async global->LDS when available). 256 threads (8 waves) per block.
// ---------------------------------------------------------------------------
__global__ __launch_bounds__(256) void k_fused(
    const float* __restrict__ X, const float* __restrict__ Wf,
    const float* __restrict__ cbias, const float* __restrict__ gamma,
    const float* __restrict__ beta, float* __restrict__ out) {
  __shared__ float sB[4 * 16 * 32 * 2];       // B-matrix, WMMA lane layout (16 KB)
  __shared__ float sA[8][2][16 * APAD];       // per-wave double-buffered A tiles

  const int tid  = threadIdx.x;
  const int lane = tid & 31;
  const int wave = tid >> 5;

  // Stage W' = Wf[:64] + I into exact B-matrix layout:
  // pair idx = (nt*16+kk)*32 + l :  b0 = W'[kk*4 + 2*(l>=16)][nt*16 + l%16]
  for (int idx = tid; idx < 4 * 16 * 32; idx += 256) {
    const int l  = idx & 31;
    const int kk = (idx >> 5) & 15;
    const int nt = idx >> 9;
    const int n  = nt * 16 + (l & 15);
    const int k0 = kk * 4 + ((l >> 4) << 1);
    sB[idx * 2]     = Wf[k0 * Dn + n]       + ((k0 == n)     ? 1.f : 0.f);
    sB[idx * 2 + 1] = Wf[(k0 + 1) * Dn + n] + ((k0 + 1 == n) ? 1.f : 0.f);
  }
  __syncthreads();

  const int m    = lane & 15;
  const int koff = (lane >> 4) << 1;

  // A wave's TPW tiles are 64 consecutive rows starting at a multiple of 64,
  // and 2048 % 64 == 0, so all tiles share one batch index b.
  const size_t tile0 = ((size_t)blockIdx.x * 8 + wave) * TPW;
  const int b = (int)((tile0 * 16) >> 11);

  float gam[4], bet[4], ci[4];
#pragma unroll
  for (int nt = 0; nt < 4; ++nt) {
    gam[nt] = gamma[nt * 16 + m];
    bet[nt] = beta[nt * 16 + m];
    ci[nt]  = cbias[b * Dn + nt * 16 + m];
  }

#if HAVE_ASYNC_LDS
  // Async global->LDS staging: data bypasses VGPRs, tracked by ASYNCcnt.
  auto stage_async = [&](int it) {
    const float* src = X + (tile0 + it) * 16 * Dn;
    float* dst = &sA[wave][it & 1][0];
#pragma unroll
    for (int r = 0; r < 8; ++r) {
      const int idx = r * 32 + lane;       // contiguous 512B per iteration
      const int row = idx >> 4, c4 = idx & 15;
      __builtin_amdgcn_global_load_async_to_lds_b128(
          (gv4i*)(src + idx * 4),
          (lv4i*)(dst + row * APAD + c4 * 4),
          0, 0);
    }
  };
  stage_async(0);
#else
  // Fallback: pipeline next tile through VGPRs, ds_store before compute.
  float4 pre[8];
  auto load_tile = [&](int it) {
    const float4* src = (const float4*)(X + (tile0 + it) * 16 * Dn);
#pragma unroll
    for (int r = 0; r < 8; ++r) pre[r] = src[r * 32 + lane];
  };
  auto store_tile = [&](int buf) {
    float* dst = &sA[wave][buf][0];
#pragma unroll
    for (int r = 0; r < 8; ++r) {
      const int idx = r * 32 + lane;
      const int row = idx >> 4, c4 = idx & 15;
      float* p = dst + row * APAD + c4 * 4;
      p[0] = pre[r].x; p[1] = pre[r].y; p[2] = pre[r].z; p[3] = pre[r].w;
    }
  };
  load_tile(0);
#endif

  for (int it = 0; it < TPW; ++it) {
    const int buf = it & 1;
    const size_t row0 = (tile0 + it) * 16;

#if HAVE_ASYNC_LDS
    if (it + 1 < TPW) {
      stage_async(it + 1);   // 8 more async ops in flight for the next buffer
      WAIT_ASYNC(8);         // previous 8 (current buffer) are complete
    } else {
      WAIT_ASYNC(0);
    }
    asm volatile("" ::: "memory");  // keep LDS reads below the wait
#else
    store_tile(buf);                 // ds ops are wave-ordered vs the reads below
    if (it + 1 < TPW) load_tile(it + 1);
#endif

    // A tile -> 32 VGPRs in WMMA A layout:
    // pair kk: a.x = X[m][4kk+koff], a.y = X[m][4kk+koff+1], koff=2*(lane>=16)
    const float* arow = &sA[wave][buf][0] + m * APAD + koff;
    v2f a[16];
#pragma unroll
    for (int kk = 0; kk < 16; ++kk) {
      a[kk].x = arow[kk * 4];
      a[kk].y = arow[kk * 4 + 1];
    }

    v8f acc[4];
#pragma unroll
    for (int nt = 0; nt < 4; ++nt) {
      const float c0 = ci[nt];
      v8f c = {c0, c0, c0, c0, c0, c0, c0, c0};
#pragma unroll
      for (int kk = 0; kk < 16; ++kk) {
        v2f bv = ((const v2f*)sB)[(nt * 16 + kk) * 32 + lane];
        c = __builtin_amdgcn_wmma_f32_16x16x4_f32(
            false, a[kk], false, bv, (short)0, c, false, false);
      }
      acc[nt] = c;
    }

    // LayerNorm over the 64 columns of each row.
    // D layout: VGPR v -> row v (lanes 0-15) / v+8 (lanes 16-31), col = nt*16+m.
    float psum[8], psq[8];
#pragma unroll
    for (int v = 0; v < 8; ++v) {
      psum[v] = acc[0][v] + acc[1][v] + acc[2][v] + acc[3][v];
      psq[v]  = acc[0][v] * acc[0][v] + acc[1][v] * acc[1][v] +
                acc[2][v] * acc[2][v] + acc[3][v] * acc[3][v];
    }
#pragma unroll
    for (int off = 1; off < 16; off <<= 1) {
#pragma unroll
      for (int v = 0; v < 8; ++v) {
        psum[v] += __shfl_xor(psum[v], off, 32);
        psq[v]  += __shfl_xor(psq[v],  off, 32);
      }
    }

    const int rbase = (lane >> 4) << 3;  // 0 or 8
#pragma unroll
    for (int v = 0; v < 8; ++v) {
      const float mu  = psum[v] * (1.f / 64.f);
      const float var = psq[v] * (1.f / 64.f) - mu * mu;
      const float rs  = rsqrtf(var + 1e-5f);
      float* orow = out + (row0 + (size_t)(v + rbase)) * Dn + m;
#pragma unroll
      for (int nt = 0; nt < 4; ++nt) {
        const float y = acc[nt][v];
        // Output is never re-read: non-temporal, don't pollute L2.
        __builtin_nontemporal_store(gam[nt] * (y - mu) * rs + bet[nt],
                                    orow + nt * 16);
      }
    }
  }
}

// ---------------------------------------------------------------------------
extern "C" void kernel_launch(void* const* d_in, const int* in_sizes, int n_in,
                              void* d_out, int out_size, void* d_ws,
                              size_t ws_size, hipStream_t stream) {
  const float* latent = (const float*)d_in[0];
  const float* Wu     = (const float*)d_in[3];
  const float* Ws     = (const float*)d_in[5];
  const float* lw     = (const float*)d_in[7];
  const float* Wf     = (const float*)d_in[8];
  const float* bf     = (const float*)d_in[9];
  const float* gamma  = (const float*)d_in[10];
  const float* beta   = (const float*)d_in[11];
  float* out = (float*)d_out;

  float* ws    = (float*)d_ws;
  float* u     = ws;                    // 512*64
  float* cbias = ws + 32768;            // 512*64
  float* e     = ws + 65536;            // 512
  float* mean0 = e + 512;
  float* mean1 = mean0 + 512;
  float* invD  = mean1 + 512;
  float* lc0   = invD + 512;
  float* lc1   = lc0 + 512;
  float* ac    = lc1 + 512;
  int*   fb    = (int*)(ac + 512);

  k_mean<<<Bn, 256, 0, stream>>>(latent, u);
  k_pairwise<<<1, Bn, 0, stream>>>(u, Wu, Ws, e, mean0, mean1, invD, lc0, lc1,
                                   ac, fb);
  k_levels<<<Bn, Dn, 0, stream>>>(u, e, mean0, mean1, invD, lc0, lc1, ac, fb,
                                  lw, Wf, bf, cbias);
  k_fused<<<(Bn * Tn) / (128 * TPW), 256, 0, stream>>>(latent, Wf, cbias,
                                                       gamma, beta, out);
}